// TopoLoss_61675730370737
// MI455X (gfx1250) — compile-verified
//
#include <hip/hip_runtime.h>
#include <cstdint>

#define BB   16
#define HH   512
#define WW   512
#define HWs  (HH * WW)           // 262144
#define KK   512                 // max regions per image per mask
#define NPIX (BB * HWs)          // 4,194,304
#define SENT 0x7fffffff

typedef float v2f __attribute__((ext_vector_type(2)));
typedef float v8f __attribute__((ext_vector_type(8)));

// ---------------- union-find (lock-free, min-index roots) ----------------
__device__ inline int uf_find(const int* L, int x) {
  volatile const int* vL = (volatile const int*)L;
  int p = vL[x];
  while (p != x) { x = p; p = vL[x]; }
  return x;
}

__device__ inline void uf_union(int* L, int a, int b) {
  a = uf_find(L, a);
  b = uf_find(L, b);
  while (a != b) {
    if (a < b) { int t = a; a = b; b = t; }   // a > b
    int old = atomicMin(&L[a], b);
    if (old == a) break;                       // linked a -> b
    // L[a] was already 'old' (< a); now must union(old, b)
    int hi = old > b ? old : b;
    int lo = old > b ? b : old;
    a = hi; b = lo;
  }
}

__device__ inline int bsearch_i(const int* arr, int n, int key) {
  int lo = 0, hi = n - 1;
  while (lo <= hi) {
    int mid = (lo + hi) >> 1;
    int v = arr[mid];
    if (v == key) return mid;
    if (v < key) lo = mid + 1; else hi = mid - 1;
  }
  return -1;
}

// ---------------- kernels ----------------
__global__ void k_zero(int* z, int n) {
  int i = blockIdx.x * blockDim.x + threadIdx.x;
  if (i < n) z[i] = 0;
}

__global__ void k_init(const float* __restrict__ in, const float* __restrict__ tg,
                       int* labP, int* labG, uint8_t* pm, uint8_t* gm) {
  int g = blockIdx.x * blockDim.x + threadIdx.x;   // < NPIX
  if (g >= NPIX) return;
  int p = g & (HWs - 1);
  bool P = in[g] > 0.0f;      // sigmoid(x) > 0.5  <=>  x > 0
  bool G = tg[g] > 0.5f;
  pm[g] = P ? 1 : 0;
  gm[g] = G ? 1 : 0;
  labP[g] = P ? p : SENT;
  labG[g] = G ? p : SENT;
}

__global__ void k_merge(int* labP, int* labG,
                        const uint8_t* __restrict__ pm, const uint8_t* __restrict__ gm) {
  int g = blockIdx.x * blockDim.x + threadIdx.x;   // < 2*NPIX
  if (g >= 2 * NPIX) return;
  int which = g / NPIX;
  int gi = g - which * NPIX;
  const uint8_t* M = which ? gm : pm;
  if (!M[gi]) return;
  int img = gi / HWs, p = gi & (HWs - 1);
  int r = p / WW, c = p & (WW - 1);
  int* Li = (which ? labG : labP) + img * HWs;
  const uint8_t* Mi = M + img * HWs;
  // half of the 8-neighborhood (each undirected edge once)
  if (c > 0 && Mi[p - 1]) uf_union(Li, p, p - 1);
  if (r > 0) {
    if (c > 0 && Mi[p - WW - 1]) uf_union(Li, p, p - WW - 1);
    if (Mi[p - WW]) uf_union(Li, p, p - WW);
    if (c < WW - 1 && Mi[p - WW + 1]) uf_union(Li, p, p - WW + 1);
  }
}

__global__ void k_flatten(int* labP, int* labG,
                          const uint8_t* __restrict__ pm, const uint8_t* __restrict__ gm) {
  int g = blockIdx.x * blockDim.x + threadIdx.x;
  if (g >= 2 * NPIX) return;
  int which = g / NPIX;
  int gi = g - which * NPIX;
  const uint8_t* M = which ? gm : pm;
  if (!M[gi]) return;
  int img = gi / HWs, p = gi & (HWs - 1);
  int* Li = (which ? labG : labP) + img * HWs;
  Li[p] = uf_find(Li, p);
}

__global__ void k_collect(const int* __restrict__ labP, const int* __restrict__ labG,
                          const uint8_t* __restrict__ pm, const uint8_t* __restrict__ gm,
                          int* cnt, int* rootsRaw) {
  int g = blockIdx.x * blockDim.x + threadIdx.x;
  if (g >= 2 * NPIX) return;
  int which = g / NPIX;
  int gi = g - which * NPIX;
  const uint8_t* M = which ? gm : pm;
  if (!M[gi]) return;
  int img = gi / HWs, p = gi & (HWs - 1);
  const int* Li = (which ? labG : labP) + img * HWs;
  if (Li[p] != p) return;                          // not a root
  int b = which * BB + img;
  int slot = atomicAdd(&cnt[b], 1);
  if (slot < KK) rootsRaw[b * KK + slot] = p;
}

__global__ void k_sort(int* cnt, const int* __restrict__ rootsRaw, int* rootsSorted) {
  __shared__ int s[KK];
  int b = blockIdx.x;                              // 0..2*BB-1
  int n = cnt[b]; if (n > KK) n = KK;
  int t = threadIdx.x;
  s[t] = (t < n) ? rootsRaw[b * KK + t] : SENT;
  __syncthreads();
  int key = s[t];
  int rank = 0;
  for (int j = 0; j < KK; ++j) {
    int v = s[j];
    rank += (v < key) || (v == key && j < t);
  }
  rootsSorted[b * KK + rank] = key;
  if (t == 0) cnt[b] = n;                          // clamp
}

__global__ void k_stats(const int* __restrict__ labP, const int* __restrict__ labG,
                        const uint8_t* __restrict__ pm, const uint8_t* __restrict__ gm,
                        const int* __restrict__ cnt, const int* __restrict__ rootsSorted,
                        int* area, int* sumr, int* sumc, int* ov) {
  int g = blockIdx.x * blockDim.x + threadIdx.x;
  if (g >= 2 * NPIX) return;
  int which = g / NPIX;
  int gi = g - which * NPIX;
  const uint8_t* M = which ? gm : pm;
  if (!M[gi]) return;
  int img = gi / HWs, p = gi & (HWs - 1);
  const int* Li = (which ? labG : labP) + img * HWs;
  int b = which * BB + img;
  int cid = bsearch_i(rootsSorted + b * KK, cnt[b], Li[p]);
  if (cid < 0) return;
  atomicAdd(&area[b * KK + cid], 1);
  atomicAdd(&sumr[b * KK + cid], p / WW);          // exact integer sums -> deterministic
  atomicAdd(&sumc[b * KK + cid], p & (WW - 1));
  // overlap flags: pred region containing any gt pixel / gt region containing any pred pixel
  const uint8_t* other = which ? pm : gm;
  if (other[gi]) ov[b * KK + cid] = 1;             // idempotent store
}

__global__ void k_centroid(const int* __restrict__ cnt, const int* __restrict__ area,
                           const int* __restrict__ sumr, const int* __restrict__ sumc,
                           float* cr, float* cc, float* nrm) {
  int id = blockIdx.x * blockDim.x + threadIdx.x;  // < 2*BB*KK
  if (id >= 2 * BB * KK) return;
  int b = id / KK, s = id - b * KK;
  if (s >= cnt[b]) return;                         // padding stays zero (pre-zeroed)
  float A = fmaxf((float)area[id], 1.0f);
  float r = (float)sumr[id] / A;
  float c = (float)sumc[id] / A;
  cr[id] = r; cc[id] = c; nrm[id] = r * r + c * c;
}

// d2[p][g] = |P|^2 + |G|^2 - 2 P.G  via V_WMMA_F32_16X16X4_F32 (K=2 used, K=3..4 zero)
__global__ void k_dist(const float* __restrict__ cr, const float* __restrict__ cc,
                       const float* __restrict__ nrm, const int* __restrict__ cnt,
                       float* __restrict__ dist) {
  int img = blockIdx.x;
  int cntP = cnt[img];
  int cntG = cnt[BB + img];
  if (cntP == 0 || cntG == 0) return;
  int nTp = (cntP + 15) >> 4, nTg = (cntG + 15) >> 4;
  int wave = threadIdx.x >> 5, lane = threadIdx.x & 31;
  const float* pr = cr + img * KK;
  const float* pc = cc + img * KK;
  const float* pn = nrm + img * KK;
  const float* gr = cr + (BB + img) * KK;
  const float* gc = cc + (BB + img) * KK;
  const float* gn = nrm + (BB + img) * KK;
  float* D = dist + (size_t)img * KK * KK;
  int nTiles = nTp * nTg;
  for (int tile = wave; tile < nTiles; tile += 8) {
    int mt = tile / nTg, nt = tile - mt * nTg;
    int m = mt * 16 + (lane & 15);
    int n = nt * 16 + (lane & 15);
    // A (16x4): lanes 0-15 hold K0,K1 = (row, col); lanes 16-31 hold K2,K3 = 0
    float ax = 0.f, ay = 0.f, bx = 0.f, by = 0.f;
    if (lane < 16) { ax = pr[m]; ay = pc[m]; bx = gr[n]; by = gc[n]; }
    v2f a = {ax, ay};
    v2f bm = {bx, by};
    v8f c = {};
    c = __builtin_amdgcn_wmma_f32_16x16x4_f32(false, a, false, bm,
                                              (short)0, c, false, false);
    float gnv = gn[n];
    bool nvalid = n < cntG;
#pragma unroll
    for (int i = 0; i < 8; ++i) {
      int mm = mt * 16 + i + ((lane >= 16) ? 8 : 0);
      float d2 = pn[mm] + gnv - 2.0f * c[i];
      if (!nvalid) d2 = 3.0e38f;
      D[(size_t)mm * KK + n] = d2;
    }
  }
}

__global__ void k_match(const int* __restrict__ cnt, const int* __restrict__ area,
                        const int* __restrict__ ov, const float* __restrict__ dist,
                        float* e_eff, int* nnArr, float* e_gt) {
  int img = blockIdx.x;
  int t = threadIdx.x;                             // blockDim = KK
  int cntP = cnt[img];
  int cntG = cnt[BB + img];
  float* eE = e_eff + img * KK;
  int*   NN = nnArr + img * KK;
  if (t < cntP) {
    if (cntG > 0) {
      const float* row = dist + (size_t)img * KK * KK + (size_t)t * KK;
      float best = row[0]; int bi = 0;
      for (int n = 1; n < cntG; ++n) {
        float v = row[n];
        if (v < best) { best = v; bi = n; }        // first-min, like jnp.argmin
      }
      float pA = fmaxf((float)area[img * KK + t], 1.0f);
      float gA = (float)area[(BB + img) * KK + bi];
      float e = fminf(fabsf(1.0f - expf(1.0f - gA / pA)), 1.0f);
      bool ok = ov[img * KK + t] != 0;             // pred overlaps gt, has_gt true
      eE[t] = ok ? e : 0.0f;
      NN[t] = bi;
    } else { eE[t] = 0.0f; NN[t] = -1; }
  }
  __syncthreads();
  if (t < cntG) {
    float m = 0.0f;
    for (int p = 0; p < cntP; ++p)
      if (NN[p] == t) m = fmaxf(m, eE[p]);
    e_gt[img * KK + t] = m;
  }
}

__global__ void k_pixel(const float* __restrict__ in,
                        const uint8_t* __restrict__ pm, const uint8_t* __restrict__ gm,
                        const int* __restrict__ labP, const int* __restrict__ labG,
                        const int* __restrict__ cnt, const int* __restrict__ rootsSorted,
                        const int* __restrict__ ov, const float* __restrict__ e_eff,
                        const float* __restrict__ e_gt,
                        float* partL, float* partP, float* partN) {
  __shared__ float sL[256], sP[256], sN[256];
  int t = threadIdx.x;
  int g = blockIdx.x * 256 + t;                    // < NPIX
  int img = g / HWs;
  float score = 1.0f / (1.0f + expf(-in[g]));
  bool P = pm[g] != 0, G = gm[g] != 0;
  int cntP = cnt[img], cntG = cnt[BB + img];
  float err = 0.0f;
  if (P) {
    if (cntG == 0) err = 1.0f;
    else {
      int cid = bsearch_i(rootsSorted + img * KK, cntP, labP[g]);
      if (cid >= 0) {
        if (!ov[img * KK + cid]) err = 1.0f;                 // pred region w/o overlap
        else if (!G) err = fmaxf(err, e_eff[img * KK + cid]); // pred pixel, gt!=1
      }
    }
  }
  if (G) {
    int gcid = bsearch_i(rootsSorted + (BB + img) * KK, cntG, labG[g]);
    if (gcid >= 0) {
      if (!P) err = fmaxf(err, e_gt[img * KK + gcid]);       // gt pixel, pred<1
      if (!ov[(BB + img) * KK + gcid]) err = 1.0f;           // gt region w/o overlap (override)
    }
  }
  float pe = err;
  float lm = (1.0f / (1.0f + expf(-(score + 1.0f) * pe)) - 0.5f) * 2.0f;
  sL[t] = lm; sP[t] = pe; sN[t] = (pe != 0.0f) ? 1.0f : 0.0f;
  __syncthreads();
  for (int s = 128; s > 0; s >>= 1) {
    if (t < s) { sL[t] += sL[t + s]; sP[t] += sP[t + s]; sN[t] += sN[t + s]; }
    __syncthreads();
  }
  if (t == 0) { partL[blockIdx.x] = sL[0]; partP[blockIdx.x] = sP[0]; partN[blockIdx.x] = sN[0]; }
}

__global__ void k_final(const float* __restrict__ pL, const float* __restrict__ pP,
                        const float* __restrict__ pN, int nb, float* out) {
  __shared__ float a[256], b[256], c[256];
  int t = threadIdx.x;
  float x = 0.f, y = 0.f, z = 0.f;
  for (int i = t; i < nb; i += 256) { x += pL[i]; y += pP[i]; z += pN[i]; }
  a[t] = x; b[t] = y; c[t] = z;
  __syncthreads();
  for (int s = 128; s > 0; s >>= 1) {
    if (t < s) { a[t] += a[t + s]; b[t] += b[t + s]; c[t] += c[t + s]; }
    __syncthreads();
  }
  if (t == 0) {
    const float N = (float)NPIX;
    out[0] = a[0] / N;            // loss
    out[1] = 1.0f - b[0] / N;     // m1
    out[2] = 1.0f - b[0] / c[0];  // m2
  }
}

// ---------------- launch ----------------
extern "C" void kernel_launch(void* const* d_in, const int* in_sizes, int n_in,
                              void* d_out, int out_size, void* d_ws, size_t ws_size,
                              hipStream_t stream) {
  const float* inp = (const float*)d_in[0];
  const float* tgt = (const float*)d_in[1];
  float* out = (float*)d_out;

  uint8_t* ws = (uint8_t*)d_ws;
  size_t off = 0;
  auto take = [&](size_t bytes) -> void* {
    void* p = ws + off;
    off = (off + bytes + 255) & ~(size_t)255;
    return p;
  };
  int*      labP = (int*)take((size_t)NPIX * 4);
  int*      labG = (int*)take((size_t)NPIX * 4);
  uint8_t*  pm   = (uint8_t*)take(NPIX);
  uint8_t*  gm   = (uint8_t*)take(NPIX);
  float*    dist = (float*)take((size_t)BB * KK * KK * 4);
  const int NB   = NPIX / 256;                       // 16384 blocks
  float*    partL = (float*)take((size_t)NB * 4);
  float*    partP = (float*)take((size_t)NB * 4);
  float*    partN = (float*)take((size_t)NB * 4);

  // contiguous zero zone
  const int BK = BB * KK;
  const int ZWORDS = 32 + 2 * BK * 9 + BK * 3;
  int* zone        = (int*)take((size_t)ZWORDS * 4);
  int* cnt         = zone;                 // 32
  int* rootsRaw    = zone + 32;            // 2*BK
  int* rootsSorted = rootsRaw + 2 * BK;    // 2*BK
  int* area        = rootsSorted + 2 * BK; // 2*BK
  int* sumr        = area + 2 * BK;        // 2*BK
  int* sumc        = sumr + 2 * BK;        // 2*BK
  int* ov          = sumc + 2 * BK;        // 2*BK
  float* cr        = (float*)(ov + 2 * BK);      // 2*BK
  float* cc        = cr + 2 * BK;                // 2*BK
  float* nrm       = cc + 2 * BK;                // 2*BK
  float* e_eff     = nrm + 2 * BK;               // BK
  int*   nnArr     = (int*)(e_eff + BK);         // BK
  float* e_gt      = (float*)(nnArr + BK);       // BK

  k_zero<<<(ZWORDS + 255) / 256, 256, 0, stream>>>(zone, ZWORDS);
  k_init<<<NPIX / 256, 256, 0, stream>>>(inp, tgt, labP, labG, pm, gm);
  k_merge<<<2 * NPIX / 256, 256, 0, stream>>>(labP, labG, pm, gm);
  k_flatten<<<2 * NPIX / 256, 256, 0, stream>>>(labP, labG, pm, gm);
  k_collect<<<2 * NPIX / 256, 256, 0, stream>>>(labP, labG, pm, gm, cnt, rootsRaw);
  k_sort<<<2 * BB, KK, 0, stream>>>(cnt, rootsRaw, rootsSorted);
  k_stats<<<2 * NPIX / 256, 256, 0, stream>>>(labP, labG, pm, gm, cnt, rootsSorted,
                                              area, sumr, sumc, ov);
  k_centroid<<<(2 * BK + 255) / 256, 256, 0, stream>>>(cnt, area, sumr, sumc, cr, cc, nrm);
  k_dist<<<BB, 256, 0, stream>>>(cr, cc, nrm, cnt, dist);
  k_match<<<BB, KK, 0, stream>>>(cnt, area, ov, dist, e_eff, nnArr, e_gt);
  k_pixel<<<NB, 256, 0, stream>>>(inp, pm, gm, labP, labG, cnt, rootsSorted,
                                  ov, e_eff, e_gt, partL, partP, partN);
  k_final<<<1, 256, 0, stream>>>(partL, partP, partN, NB, out);
}